// ToyDGN_69647189672208
// MI455X (gfx1250) — compile-verified
//
#include <hip/hip_runtime.h>

// ---------------------------------------------------------------------------
// ToyDGN (GraphSAGE x4 + projection) for MI455X / gfx1250, wave32.
//   Phase 1: degree count (native f32 atomics) + reciprocal
//   Phase 2 (x4): edge scatter-mean (float4 gather + global_atomic_add_f32)
//                 fused dual-GEMM  h' = (agg*rdeg)@Wl + h@Wr + bl  via
//                 V_WMMA_F32_16X16X4_F32, weights pre-swizzled in LDS so each
//                 B fragment is one aligned, bank-conflict-free ds_load_b64
//   Phase 3: projection hcat[N,256] @ Wp[256,32] + bp via same WMMA path
// Workspace: rdeg[N] | agg[N*64] | hcat[N*256]  (~128.5 MB)
// ---------------------------------------------------------------------------

typedef float v2f __attribute__((ext_vector_type(2)));
typedef float v8f __attribute__((ext_vector_type(8)));

__device__ __forceinline__ v2f mk2(float a, float b) { v2f r; r.x = a; r.y = b; return r; }

#define WMMA4(a, b, c) __builtin_amdgcn_wmma_f32_16x16x4_f32( \
    false, (a), false, (b), (short)0, (c), false, false)

__device__ __forceinline__ void atomAddF32(float* p, float v) {
    unsafeAtomicAdd(p, v);   // native global_atomic_add_f32 (no CAS loop)
}

constexpr int D  = 64;
constexpr int L  = 4;
constexpr int DT = 32;
constexpr int LD = L * D;        // 256
constexpr int SP_W = 160;        // LDS floats per k-pair row (64x64 W): 128 + 32 pad
constexpr int SP_P = 96;         // LDS floats per k-pair row (256x32 Wp): 64 + 32 pad

// ---------------- degree kernels ----------------
__global__ void __launch_bounds__(256)
deg_count_kernel(const int* __restrict__ ei, long long n_edges, float* __restrict__ deg) {
    long long e = (long long)blockIdx.x * blockDim.x + threadIdx.x;
    if (e >= n_edges) return;
    int dst = ei[n_edges + e];
    atomAddF32(&deg[dst], 1.0f);
}

__global__ void __launch_bounds__(256)
deg_recip_kernel(float* __restrict__ deg, int n) {
    int i = blockIdx.x * blockDim.x + threadIdx.x;
    if (i < n) deg[i] = 1.0f / fmaxf(deg[i], 1.0f);
}

// ---------------- edge scatter: agg[dst] += h[src] ----------------
// 16 lanes per edge, 4 features each (float4 gather + 4 native f32 atomics).
__global__ void __launch_bounds__(256)
edge_scatter_kernel(const int* __restrict__ ei, long long n_edges,
                    const float* __restrict__ hin, int sIn,
                    float* __restrict__ agg) {
    long long t = (long long)blockIdx.x * blockDim.x + threadIdx.x;
    long long e = t >> 4;
    int c = (int)(t & 15);
    if (e >= n_edges) return;
    int src = ei[e];
    int dst = ei[n_edges + e];
    const float4 v = *(const float4*)(hin + (size_t)src * sIn + c * 4);
    float* d = agg + (size_t)dst * D + c * 4;
    atomAddF32(d + 0, v.x);
    atomAddF32(d + 1, v.y);
    atomAddF32(d + 2, v.z);
    atomAddF32(d + 3, v.w);
}

// ---------------- fused SAGE layer GEMM ----------------
// hout[r, 0:64] = (agg[r]*rdeg[r]) @ Wl + hin[r] @ Wr + bl
// One wave per 16-row tile; 4 N-tiles of 16; K=64 in steps of 4.
// LDS B layout: sW[p*SP_W + 2n + j] = W[2p+j][n]  -> fragment = 1x ds_load_b64.
__global__ void __launch_bounds__(256)
layer_gemm_kernel(const float* __restrict__ agg, const float* __restrict__ rdeg,
                  const float* __restrict__ hin, int sIn,
                  const float* __restrict__ Wl, const float* __restrict__ Wr,
                  const float* __restrict__ bl,
                  float* __restrict__ hout, int sOut, int n) {
    __shared__ __align__(16) float sWl[(D / 2) * SP_W];   // 20 KB
    __shared__ __align__(16) float sWr[(D / 2) * SP_W];   // 20 KB
    for (int i = threadIdx.x; i < (D / 2) * D; i += 256) {  // 2048 k-pair x n entries
        const int p = i >> 6;          // k-pair
        const int c = i & 63;          // n
        const int g0 = (2 * p) * D + c;
        *(float2*)&sWl[p * SP_W + 2 * c] = make_float2(Wl[g0], Wl[g0 + D]);
        *(float2*)&sWr[p * SP_W + 2 * c] = make_float2(Wr[g0], Wr[g0 + D]);
    }
    __syncthreads();

    const int lane = threadIdx.x & 31;
    const int wave = threadIdx.x >> 5;
    const int tile = blockIdx.x * 8 + wave;
    const int tiles = (n + 15) >> 4;
    if (tile >= tiles) return;

    const int row0 = tile << 4;
    const int m    = lane & 15;       // M (for A) / N (for B,D) index within tile
    const int half = lane >> 4;       // 0: K=0,1   1: K=2,3
    const int kg   = half << 1;

    int row = row0 + m; if (row >= n) row = n - 1;
    const float rd = rdeg[row];
    const float* aRow = agg + (size_t)row * D;
    const float* hRow = hin + (size_t)row * sIn;

    const v2f* fl = (const v2f*)sWl;  // SP_W/2 = 80 v2f per k-pair row
    const v2f* fr = (const v2f*)sWr;

    v8f c0 = {}, c1 = {}, c2 = {}, c3 = {};

#pragma unroll 4
    for (int k0 = 0; k0 < D; k0 += 4) {
        const int ka = k0 + kg;
        const float2 a2 = *(const float2*)(aRow + ka);
        const float2 h2 = *(const float2*)(hRow + ka);
        const v2f aA = mk2(a2.x * rd, a2.y * rd);
        const v2f aH = mk2(h2.x, h2.y);
        const int fb = (ka >> 1) * (SP_W / 2) + m;   // fragment base (v2f units)
        c0 = WMMA4(aA, fl[fb],      c0);
        c0 = WMMA4(aH, fr[fb],      c0);
        c1 = WMMA4(aA, fl[fb + 16], c1);
        c1 = WMMA4(aH, fr[fb + 16], c1);
        c2 = WMMA4(aA, fl[fb + 32], c2);
        c2 = WMMA4(aH, fr[fb + 32], c2);
        c3 = WMMA4(aA, fl[fb + 48], c3);
        c3 = WMMA4(aH, fr[fb + 48], c3);
    }

    const float b0 = bl[m], b1 = bl[16 + m], b2 = bl[32 + m], b3 = bl[48 + m];
    const int rbase = row0 + half * 8;        // D layout: vgpr v -> M = v + 8*half
#pragma unroll
    for (int v = 0; v < 8; ++v) {
        const int r = rbase + v;
        if (r < n) {
            float* o = hout + (size_t)r * sOut;
            o[m]      = c0[v] + b0;
            o[16 + m] = c1[v] + b1;
            o[32 + m] = c2[v] + b2;
            o[48 + m] = c3[v] + b3;
        }
    }
}

// ---------------- projection: out = hcat @ Wp + bp ----------------
__global__ void __launch_bounds__(256)
proj_gemm_kernel(const float* __restrict__ hcat, const float* __restrict__ Wp,
                 const float* __restrict__ bp, float* __restrict__ out, int n) {
    __shared__ __align__(16) float sW[(LD / 2) * SP_P];   // 48 KB
    for (int i = threadIdx.x; i < (LD / 2) * DT; i += 256) {  // 4096 entries
        const int p = i >> 5;          // k-pair
        const int c = i & 31;          // n
        const int g0 = (2 * p) * DT + c;
        *(float2*)&sW[p * SP_P + 2 * c] = make_float2(Wp[g0], Wp[g0 + DT]);
    }
    __syncthreads();

    const int lane = threadIdx.x & 31;
    const int wave = threadIdx.x >> 5;
    const int tile = blockIdx.x * 8 + wave;
    const int tiles = (n + 15) >> 4;
    if (tile >= tiles) return;

    const int row0 = tile << 4;
    const int m    = lane & 15;
    const int half = lane >> 4;
    const int kg   = half << 1;

    int row = row0 + m; if (row >= n) row = n - 1;
    const float* hRow = hcat + (size_t)row * LD;
    const v2f* fp = (const v2f*)sW;   // SP_P/2 = 48 v2f per k-pair row

    v8f c0 = {}, c1 = {};
#pragma unroll 8
    for (int k0 = 0; k0 < LD; k0 += 4) {
        const int ka = k0 + kg;
        const float2 h2 = *(const float2*)(hRow + ka);
        const v2f aH = mk2(h2.x, h2.y);
        const int fb = (ka >> 1) * (SP_P / 2) + m;
        c0 = WMMA4(aH, fp[fb],      c0);
        c1 = WMMA4(aH, fp[fb + 16], c1);
    }

    const float b0 = bp[m], b1 = bp[16 + m];
    const int rbase = row0 + half * 8;
#pragma unroll
    for (int v = 0; v < 8; ++v) {
        const int r = rbase + v;
        if (r < n) {
            out[(size_t)r * DT + m]      = c0[v] + b0;
            out[(size_t)r * DT + 16 + m] = c1[v] + b1;
        }
    }
}

// ---------------------------------------------------------------------------
extern "C" void kernel_launch(void* const* d_in, const int* in_sizes, int n_in,
                              void* d_out, int out_size, void* d_ws, size_t ws_size,
                              hipStream_t stream) {
    const float* x  = (const float*)d_in[0];
    const int*   ei = (const int*)d_in[1];     // edge_index (harness integer convention)
    const float* Wl = (const float*)d_in[2];   // [L,64,64]
    const float* bl = (const float*)d_in[3];   // [L,64]
    const float* Wr = (const float*)d_in[4];   // [L,64,64]
    const float* Wp = (const float*)d_in[5];   // [256,32]
    const float* bp = (const float*)d_in[6];   // [32]
    float* out = (float*)d_out;

    const int       n  = in_sizes[0] / D;              // 100000
    const long long nE = (long long)in_sizes[1] / 2;   // 1600000

    // Workspace layout (floats): rdeg[n] | agg[n*64] | hcat[n*256]
    float* ws   = (float*)d_ws;
    float* rdeg = ws;
    float* aggb = ws + (((size_t)n + 255) & ~(size_t)255);
    float* hcat = aggb + (size_t)n * D;

    // --- degrees ---
    hipMemsetAsync(rdeg, 0, sizeof(float) * (size_t)n, stream);
    {
        dim3 b(256), g((unsigned)((nE + 255) / 256));
        deg_count_kernel<<<g, b, 0, stream>>>(ei, nE, rdeg);
    }
    {
        dim3 b(256), g((unsigned)((n + 255) / 256));
        deg_recip_kernel<<<g, b, 0, stream>>>(rdeg, n);
    }

    const int tiles = (n + 15) / 16;
    const int gemmBlocks = (tiles + 7) / 8;
    const long long scatterThreads = nE * 16;
    const unsigned scatterBlocks = (unsigned)((scatterThreads + 255) / 256);

    // --- 4 SAGE layers ---
    for (int i = 0; i < L; ++i) {
        const float* hin = (i == 0) ? x : (hcat + (size_t)(i - 1) * D);
        const int    sIn = (i == 0) ? D : LD;

        hipMemsetAsync(aggb, 0, sizeof(float) * (size_t)n * D, stream);
        edge_scatter_kernel<<<scatterBlocks, 256, 0, stream>>>(ei, nE, hin, sIn, aggb);
        layer_gemm_kernel<<<gemmBlocks, 256, 0, stream>>>(
            aggb, rdeg, hin, sIn,
            Wl + (size_t)i * D * D, Wr + (size_t)i * D * D, bl + (size_t)i * D,
            hcat + (size_t)i * D, LD, n);
    }

    // --- projection ---
    proj_gemm_kernel<<<gemmBlocks, 256, 0, stream>>>(hcat, Wp, bp, out, n);
}